// VectorQuantizer_40888088658388
// MI455X (gfx1250) — compile-verified
//
#include <hip/hip_runtime.h>
#include <hip/hip_bf16.h>

typedef __attribute__((ext_vector_type(2))) float v2f;
typedef __attribute__((ext_vector_type(8))) float v8f;

#define VQ_N_EMBED 8192
#define VQ_E_DIM   32
#define VQ_TOKENS  32768

// ---------------------------------------------------------------------------
// Kernel 1: squared L2 norms of the 8192 codebook rows -> d_ws
// ---------------------------------------------------------------------------
__global__ __launch_bounds__(256) void vq_enorm_kernel(
    const float* __restrict__ emb, float* __restrict__ norms)
{
    int n = blockIdx.x * blockDim.x + threadIdx.x;
    if (n >= VQ_N_EMBED) return;
    const float4* row = (const float4*)(emb + n * VQ_E_DIM);
    float s = 0.f;
#pragma unroll
    for (int i = 0; i < VQ_E_DIM / 4; ++i) {
        float4 v = row[i];
        s += v.x * v.x + v.y * v.y + v.z * v.z + v.w * v.w;
    }
    norms[n] = s;
}

// ---------------------------------------------------------------------------
// Kernel 2: fused GEMM (V_WMMA_F32_16X16X4_F32 chain) + running argmin
//   wave -> 16 tokens; 512 N-tiles of 16 codes; exact f32 dot products.
// ---------------------------------------------------------------------------
__global__ __launch_bounds__(256) void vq_argmin_kernel(
    const float* __restrict__ z,      // [32768, 32]
    const float* __restrict__ emb,    // [8192, 32]
    const float* __restrict__ eNorm,  // [8192]
    float* __restrict__ zq,           // [32768, 32]
    int* __restrict__ outIdx)         // [32768]
{
    const int lane = threadIdx.x & 31;
    const int wave = threadIdx.x >> 5;
    const int tokenBase = blockIdx.x * 128 + wave * 16;

    // A-fragment addressing (16x4 f32): lane l -> row m=l&15, K-pair kb=(l>>4)*2
    const int mA = lane & 15;
    const int kb = (lane >> 4) * 2;

    // Preload the 8 K-step A fragments for this wave's 16 tokens (kept in VGPRs)
    v2f a[8];
    const float* zrow = z + (size_t)(tokenBase + mA) * VQ_E_DIM + kb;
#pragma unroll
    for (int s = 0; s < 8; ++s)
        a[s] = *(const v2f*)(zrow + 4 * s);

    float best[8];
    int   bidx[8];
#pragma unroll
    for (int r = 0; r < 8; ++r) { best[r] = 3.402823466e38f; bidx[r] = 0; }

    const int nB = lane & 15;   // B-fragment column, same striping as A rows

#pragma unroll 2
    for (int nBase = 0; nBase < VQ_N_EMBED; nBase += 16) {
        // B-fragments: e^T[k][n] = emb[n][k], row-major float2 loads (L2 resident)
        const float* erow = emb + (size_t)(nBase + nB) * VQ_E_DIM + kb;
        v2f b[8];
#pragma unroll
        for (int s = 0; s < 8; ++s)
            b[s] = *(const v2f*)(erow + 4 * s);
        float en = eNorm[nBase + nB];

        v8f acc = {0.f, 0.f, 0.f, 0.f, 0.f, 0.f, 0.f, 0.f};
#pragma unroll
        for (int s = 0; s < 8; ++s)
            acc = __builtin_amdgcn_wmma_f32_16x16x4_f32(
                false, a[s], false, b[s], (short)0, acc, false, false);

        // distance (up to per-token constant ||z||^2): ||e||^2 - 2 z.e
#pragma unroll
        for (int r = 0; r < 8; ++r) {
            float d = __builtin_fmaf(-2.0f, acc[r], en);
            if (d < best[r]) { best[r] = d; bidx[r] = nBase + nB; }
        }
    }

    // Min+argmin reduce over the 16 lanes sharing each token
    // (xor 1,2,4,8 stays within each half-wave; tie-break on smaller index)
#pragma unroll
    for (int off = 1; off < 16; off <<= 1) {
#pragma unroll
        for (int r = 0; r < 8; ++r) {
            float ov = __shfl_xor(best[r], off, 32);
            int   oi = __shfl_xor(bidx[r], off, 32);
            if (ov < best[r] || (ov == best[r] && oi < bidx[r])) {
                best[r] = ov; bidx[r] = oi;
            }
        }
    }

    // Winner for token (tokenBase + r)     lives in lane 0
    // Winner for token (tokenBase + r + 8) lives in lane 16
#pragma unroll
    for (int r = 0; r < 8; ++r) {
        int i0 = __shfl(bidx[r], 0, 32);
        int i1 = __shfl(bidx[r], 16, 32);
        // coalesced gather: one 32-float codebook row per token, 1 dim per lane
        zq[(size_t)(tokenBase + r) * VQ_E_DIM + lane]     = emb[(size_t)i0 * VQ_E_DIM + lane];
        zq[(size_t)(tokenBase + r + 8) * VQ_E_DIM + lane] = emb[(size_t)i1 * VQ_E_DIM + lane];
        if (lane == 0) {
            outIdx[tokenBase + r]     = i0;
            outIdx[tokenBase + r + 8] = i1;
        }
    }
}

// ---------------------------------------------------------------------------
extern "C" void kernel_launch(void* const* d_in, const int* in_sizes, int n_in,
                              void* d_out, int out_size, void* d_ws, size_t ws_size,
                              hipStream_t stream)
{
    const float* z   = (const float*)d_in[0];   // hidden_states (32,32,32,32) f32
    const float* emb = (const float*)d_in[1];   // emb_weights  (8192,32) f32

    float* eNorm = (float*)d_ws;                                 // 8192 floats
    float* zq    = (float*)d_out;                                // 32768*32 floats
    int*   oidx  = (int*)((float*)d_out + (size_t)VQ_TOKENS * VQ_E_DIM); // 32768 ints

    vq_enorm_kernel<<<VQ_N_EMBED / 256, 256, 0, stream>>>(emb, eNorm);
    // 256 blocks * 8 waves * 16 tokens = 32768 tokens
    vq_argmin_kernel<<<VQ_TOKENS / 128, 256, 0, stream>>>(z, emb, eNorm, zq, oidx);
}